// AxialAttention_3745211482827
// MI455X (gfx1250) — compile-verified
//
#include <hip/hip_runtime.h>
#include <hip/hip_bf16.h>
#include <math.h>

typedef __attribute__((ext_vector_type(2))) float v2f;
typedef __attribute__((ext_vector_type(8))) float v8f;

// Problem constants (fixed by the reference)
#define NSEQ   4096        // B*T*W
#define DM     256         // d_model
#define SEQL   32          // H (axial length) == SPAN
#define NHEAD  8

// B-tile LDS layout: K-pair interleaved, stride 96 words per c-pair so the
// two 16-lane halves of a wave address disjoint bank halves (96 mod 64 == 32).
#define BSTR   96

// ws layout in floats
#define XP_OFF  0ull
#define XP_SZ   ((unsigned long long)NSEQ * 8192ull)               // xp[n][h][c]
#define ST_OFF  (XP_OFF + XP_SZ)
#define ST_SZ   ((unsigned long long)NSEQ * 512ull * 32ull)        // stacked[n][ch][i]
#define PS_OFF  (ST_OFF + ST_SZ)
#define PS_SZ   (512ull * (unsigned long long)NSEQ)                // psum[ch][n]
#define PQ_OFF  (PS_OFF + PS_SZ)
#define PQ_SZ   (512ull * (unsigned long long)NSEQ)                // psq[ch][n]
#define SC_OFF  (PQ_OFF + PQ_SZ)                                   // scale[512]
#define SH_OFF  (SC_OFF + 512ull)                                  // shift[512]

__device__ __forceinline__ v8f v8zero() {
  v8f v;
#pragma unroll
  for (int i = 0; i < 8; ++i) v[i] = 0.0f;
  return v;
}

// CDNA5 async global->LDS copy (4B per active lane, per-lane LDS scatter addr).
// Tracked by ASYNCcnt; pair with async_wait0() before consuming the LDS data.
__device__ __forceinline__ void async_g2l_b32(void* lds_dst, const void* gsrc) {
  unsigned lds_a = (unsigned)(size_t)lds_dst;   // addrspace(3) ptrtoint = LDS byte offset
  asm volatile("global_load_async_to_lds_b32 %0, %1, off"
               :: "v"(lds_a), "v"(gsrc) : "memory");
}
__device__ __forceinline__ void async_wait0() {
  asm volatile("s_wait_asynccnt 0x0" ::: "memory");
}

// ---------------------------------------------------------------------------
// Kernel 1: pack x (B,C,H,W,T) -> xp[n][h][c], n = (b*32+t)*32+w.
// Reads coalesced along t (async DMA into a transposed LDS tile),
// writes coalesced along c.
// ---------------------------------------------------------------------------
__global__ void __launch_bounds__(256) k_pack_x(const float* __restrict__ x,
                                                float* __restrict__ xp) {
  const int bi = blockIdx.x;           // (b, h, w)
  const int b = bi >> 10;
  const int h = (bi >> 5) & 31;
  const int w = bi & 31;

  __shared__ float tile[32 * 257];     // [t][c], padded stride 257

  const int t  = threadIdx.x & 31;
  const int cq = threadIdx.x >> 5;
  // x[b][c][h][w][t] flat = b*8388608 + c*32768 + h*1024 + w*32 + t
  const size_t base = (size_t)b * 8388608ull + (size_t)h * 1024ull + (size_t)w * 32ull;
#pragma unroll
  for (int cc = 0; cc < 32; ++cc) {
    const int c = cq * 32 + cc;
    async_g2l_b32(&tile[t * 257 + c], x + base + (size_t)c * 32768ull + t);
  }
  async_wait0();
  __syncthreads();

  const int c = threadIdx.x;
#pragma unroll
  for (int tr = 0; tr < 32; ++tr) {
    const size_t n = (size_t)(b * 32 + tr) * 32 + w;
    xp[n * 8192ull + h * 256 + c] = tile[tr * 257 + c];
  }
}

// ---------------------------------------------------------------------------
// Kernel 2: fused QKV GEMM + axial attention, one block per n, one wave/head.
// All GEMM-shaped contractions use v_wmma_f32_16x16x4_f32 (fp32, exact-ish).
// ---------------------------------------------------------------------------
__global__ void __launch_bounds__(256) k_attn(const float* __restrict__ xp,
                                              const float* __restrict__ Wq,
                                              const float* __restrict__ rel,
                                              float* __restrict__ stacked,
                                              float* __restrict__ psum,
                                              float* __restrict__ psq) {
  const int n    = blockIdx.x;
  const int tid  = threadIdx.x;
  const int wave = tid >> 5;           // head index (wave32)
  const int lane = tid & 31;
  const int lo   = lane & 15;
  const int hi   = lane >> 4;          // 0 | 1 : selects K pair in fragments

  __shared__ float bt[128 * BSTR];     // B tile, K-pair interleaved (P alias later)
  __shared__ float S[16896];           // qkv: [head][64][33]
  __shared__ float rl[4096];           // relative [64 rows][64 (63 used)]

  // ---- stage relative table + xp tile via async global->LDS DMA ----
  // (interleaved/transposed LDS scatter done by the per-lane VDST address)
  for (int idx = tid; idx < 64 * 63; idx += 256) {
    const int r = idx / 63, cc = idx % 63;
    async_g2l_b32(&rl[r * 64 + cc], rel + idx);
  }
  const float* xpn = xp + (size_t)n * 8192ull;
#pragma unroll
  for (int u = 0; u < 32; ++u) {
    const int idx = tid + u * 256;
    const int h = idx >> 8, c = idx & 255;       // xp is [h][c]
    async_g2l_b32(&bt[(c >> 1) * BSTR + h * 2 + (c & 1)], xpn + idx);
  }
  async_wait0();
  __syncthreads();

  // ---- Phase 1: QKV GEMM for this head: rows [wave*64, wave*64+64) ----
  v8f acc[4][2];
#pragma unroll
  for (int m = 0; m < 4; ++m)
#pragma unroll
    for (int nt = 0; nt < 2; ++nt) acc[m][nt] = v8zero();

  const int rowb = wave * 64;
  const int bbase = hi * BSTR + lo * 2;          // + ks*2*BSTR (+ nt*32)
#pragma unroll 4
  for (int ks = 0; ks < 64; ++ks) {
    const int c0 = ks * 4 + hi * 2;
    v2f a[4], bf[2];
#pragma unroll
    for (int m = 0; m < 4; ++m) {
      const float* ap = Wq + (size_t)(rowb + m * 16 + lo) * 256 + c0;
      a[m] = *reinterpret_cast<const v2f*>(ap);  // 8B aligned
    }
    const float* bp = &bt[ks * 2 * BSTR + bbase];
    bf[0] = *reinterpret_cast<const v2f*>(bp);        // hcol = lo      (ds_load_b64)
    bf[1] = *reinterpret_cast<const v2f*>(bp + 32);   // hcol = 16 + lo (ds_load_b64)
#pragma unroll
    for (int m = 0; m < 4; ++m)
#pragma unroll
      for (int nt = 0; nt < 2; ++nt)
        acc[m][nt] = __builtin_amdgcn_wmma_f32_16x16x4_f32(
            false, a[m], false, bf[nt], (short)0, acc[m][nt], false, false);
  }

  // ---- Phase 2: spill qkv to S[head][c][i] ----
  float* Sh = S + wave * 64 * 33;
#pragma unroll
  for (int m = 0; m < 4; ++m)
#pragma unroll
    for (int nt = 0; nt < 2; ++nt)
#pragma unroll
      for (int r = 0; r < 8; ++r) {
        const int c = m * 16 + r + hi * 8;
        const int i = nt * 16 + lo;
        Sh[c * 33 + i] = acc[m][nt][r];
      }
  __syncthreads();  // everyone done with bt -> reuse region as softmax scratch P

  // ---- Phase 3: qk[i,j] = sum_{c<16} q[c,i] k[c,j] via WMMA ----
  v8f qk[2][2];
#pragma unroll
  for (int ti = 0; ti < 2; ++ti)
#pragma unroll
    for (int tj = 0; tj < 2; ++tj) qk[ti][tj] = v8zero();

#pragma unroll
  for (int ks = 0; ks < 4; ++ks) {
    const int cq = ks * 4 + hi * 2;
    v2f aq[2], bk[2];
#pragma unroll
    for (int ti = 0; ti < 2; ++ti) {
      const int i = ti * 16 + lo;
      aq[ti].x = Sh[cq * 33 + i];          // A[m=i,k=c] = q[c,i]
      aq[ti].y = Sh[(cq + 1) * 33 + i];
    }
#pragma unroll
    for (int tj = 0; tj < 2; ++tj) {
      const int j = tj * 16 + lo;
      bk[tj].x = Sh[(16 + cq) * 33 + j];   // B[k=c,n=j] = k[c,j]
      bk[tj].y = Sh[(17 + cq) * 33 + j];
    }
#pragma unroll
    for (int ti = 0; ti < 2; ++ti)
#pragma unroll
      for (int tj = 0; tj < 2; ++tj)
        qk[ti][tj] = __builtin_amdgcn_wmma_f32_16x16x4_f32(
            false, aq[ti], false, bk[tj], (short)0, qk[ti][tj], false, false);
  }

  // ---- Phase 4: logits = qk + qr + kr ; write to P[head][i][j] ----
  float* Ph = bt + wave * 1056;  // 32*33 per head, lives in old bt region
#pragma unroll
  for (int ti = 0; ti < 2; ++ti)
#pragma unroll
    for (int r = 0; r < 8; ++r) {
      const int i = ti * 16 + r + hi * 8;
      float qv[16], kv[16];
#pragma unroll
      for (int c = 0; c < 16; ++c) {
        qv[c] = Sh[c * 33 + i];
        kv[c] = Sh[(16 + c) * 33 + i];
      }
#pragma unroll
      for (int tj = 0; tj < 2; ++tj) {
        const int j = tj * 16 + lo;
        const int ij = i + j;              // all_emb[c,i,j] = relative[c, i+j]
        float v = qk[ti][tj][r];
#pragma unroll
        for (int c = 0; c < 16; ++c)
          v += qv[c] * rl[c * 64 + ij] + kv[c] * rl[(16 + c) * 64 + ij];
        Ph[i * 33 + j] = v;
      }
    }
  __syncthreads();

  // ---- Phase 5: softmax over j; one lane per row i ----
  {
    float* row = Ph + lane * 33;
    float mx = -1e30f;
#pragma unroll
    for (int j = 0; j < 32; ++j) mx = fmaxf(mx, row[j]);
    float e[32], sum = 0.0f;
#pragma unroll
    for (int j = 0; j < 32; ++j) { e[j] = expf(row[j] - mx); sum += e[j]; }
    const float inv = 1.0f / sum;
#pragma unroll
    for (int j = 0; j < 32; ++j) row[j] = e[j] * inv;
  }
  __syncthreads();

  // ---- Phase 6: sv[c,i] = sum_j V[c,j] sim[i,j] via WMMA ----
  v8f sv[2][2];
#pragma unroll
  for (int ct = 0; ct < 2; ++ct)
#pragma unroll
    for (int it = 0; it < 2; ++it) sv[ct][it] = v8zero();

#pragma unroll
  for (int ks = 0; ks < 8; ++ks) {
    const int j0 = ks * 4 + hi * 2;
    v2f av[2], bs[2];
#pragma unroll
    for (int ct = 0; ct < 2; ++ct) {
      const int c = 32 + ct * 16 + lo;     // A[m=c,k=j] = v[c,j]
      av[ct].x = Sh[c * 33 + j0];
      av[ct].y = Sh[c * 33 + j0 + 1];
    }
#pragma unroll
    for (int it = 0; it < 2; ++it) {
      const int i = it * 16 + lo;          // B[k=j,n=i] = sim[i,j]
      bs[it].x = Ph[i * 33 + j0];
      bs[it].y = Ph[i * 33 + j0 + 1];
    }
#pragma unroll
    for (int ct = 0; ct < 2; ++ct)
#pragma unroll
      for (int it = 0; it < 2; ++it)
        sv[ct][it] = __builtin_amdgcn_wmma_f32_16x16x4_f32(
            false, av[ct], false, bs[it], (short)0, sv[ct][it], false, false);
  }

  // ---- Phase 7: store sv to stacked + BN partials (deterministic, no atomics)
  float* stk = stacked + (size_t)n * 16384ull;
#pragma unroll
  for (int ct = 0; ct < 2; ++ct)
#pragma unroll
    for (int r = 0; r < 8; ++r) {
      const int c  = ct * 16 + r + hi * 8;
      const int ch = wave * 64 + c * 2;    // even = sv channel
      const float v0 = sv[ct][0][r];
      const float v1 = sv[ct][1][r];
      stk[ch * 32 + lo]      = v0;
      stk[ch * 32 + 16 + lo] = v1;
      float s = v0 + v1;
      float q = v0 * v0 + v1 * v1;
#pragma unroll
      for (int m = 8; m >= 1; m >>= 1) {   // reduce within each 16-lane half
        s += __shfl_xor(s, m, 32);
        q += __shfl_xor(q, m, 32);
      }
      if (lo == 0) {
        psum[(size_t)ch * NSEQ + n] = s;
        psq [(size_t)ch * NSEQ + n] = q;
      }
    }

  // ---- Phase 8: sve[c,i] = sum_j sim[i,j] * relative[32+c, i+j] ; lane = i ----
  float pv[32];
  {
    const float* row = Ph + lane * 33;
#pragma unroll
    for (int j = 0; j < 32; ++j) pv[j] = row[j];
  }
#pragma unroll 2
  for (int c = 0; c < 32; ++c) {
    const float* rv = rl + (32 + c) * 64 + lane;
    float v = 0.0f;
#pragma unroll
    for (int j = 0; j < 32; ++j) v += pv[j] * rv[j];
    const int ch = wave * 64 + c * 2 + 1;  // odd = sve channel
    stk[ch * 32 + lane] = v;
    float s = v, q = v * v;
#pragma unroll
    for (int m = 16; m >= 1; m >>= 1) {
      s += __shfl_xor(s, m, 32);
      q += __shfl_xor(q, m, 32);
    }
    if (lane == 0) {
      psum[(size_t)ch * NSEQ + n] = s;
      psq [(size_t)ch * NSEQ + n] = q;
    }
  }
}

// ---------------------------------------------------------------------------
// Kernel 3: deterministic per-channel reduction -> BN scale/shift
// ---------------------------------------------------------------------------
__global__ void __launch_bounds__(256) k_stats(const float* __restrict__ psum,
                                               const float* __restrict__ psq,
                                               const float* __restrict__ gamma,
                                               const float* __restrict__ beta,
                                               float* __restrict__ scale,
                                               float* __restrict__ shift) {
  const int ch = blockIdx.x;
  __shared__ float rs[256], rq[256];
  float s = 0.0f, q = 0.0f;
  for (int nn = threadIdx.x; nn < NSEQ; nn += 256) {
    s += psum[(size_t)ch * NSEQ + nn];
    q += psq[(size_t)ch * NSEQ + nn];
  }
  rs[threadIdx.x] = s; rq[threadIdx.x] = q;
  __syncthreads();
  for (int off = 128; off >= 1; off >>= 1) {
    if (threadIdx.x < off) {
      rs[threadIdx.x] += rs[threadIdx.x + off];
      rq[threadIdx.x] += rq[threadIdx.x + off];
    }
    __syncthreads();
  }
  if (threadIdx.x == 0) {
    const float invN = 1.0f / (float)(NSEQ * SEQL);
    const float mean = rs[0] * invN;
    const float var  = rq[0] * invN - mean * mean;
    const float inv  = rsqrtf(var + 1e-5f);
    const float sc   = gamma[ch] * inv;
    scale[ch] = sc;
    shift[ch] = beta[ch] - mean * sc;
  }
}

// ---------------------------------------------------------------------------
// Kernel 4: normalize, pair-sum (sv+sve), transposed store to (B,C,H,W,T)
// ---------------------------------------------------------------------------
__global__ void __launch_bounds__(256) k_out(const float* __restrict__ stacked,
                                             const float* __restrict__ scale,
                                             const float* __restrict__ shift,
                                             float* __restrict__ out) {
  const int bi = blockIdx.x;          // (b, co)
  const int b  = bi >> 8;
  const int co = bi & 255;
  const int ch0 = co * 2, ch1 = ch0 + 1;
  const float s0 = scale[ch0], f0 = shift[ch0];
  const float s1 = scale[ch1], f1 = shift[ch1];

  const int t  = threadIdx.x & 31;
  const int wq = threadIdx.x >> 5;
  const size_t obase = (size_t)(b * 256 + co) * 32768ull;  // out[b][co][i][w][t]
#pragma unroll
  for (int wp = 0; wp < 4; ++wp) {
    const int w = wq * 4 + wp;
    const size_t n = (size_t)(b * 32 + t) * 32 + w;
    const float* r0 = stacked + n * 16384ull + (size_t)ch0 * 32;
    const float* r1 = r0 + 32;
#pragma unroll 4
    for (int i = 0; i < 32; ++i) {
      const float v = r0[i] * s0 + f0 + r1[i] * s1 + f1;
      out[obase + (size_t)i * 1024 + (size_t)w * 32 + t] = v;
    }
  }
}

// ---------------------------------------------------------------------------
extern "C" void kernel_launch(void* const* d_in, const int* in_sizes, int n_in,
                              void* d_out, int out_size, void* d_ws, size_t ws_size,
                              hipStream_t stream) {
  (void)in_sizes; (void)n_in; (void)out_size; (void)ws_size;
  const float* x     = (const float*)d_in[0];
  const float* Wq    = (const float*)d_in[1];
  const float* rel   = (const float*)d_in[2];
  const float* gamma = (const float*)d_in[3];
  const float* beta  = (const float*)d_in[4];
  float* ws      = (float*)d_ws;
  float* xp      = ws + XP_OFF;
  float* stacked = ws + ST_OFF;
  float* psum    = ws + PS_OFF;
  float* psq     = ws + PQ_OFF;
  float* scale   = ws + SC_OFF;
  float* shift   = ws + SH_OFF;
  float* out     = (float*)d_out;

  k_pack_x<<<4096, 256, 0, stream>>>(x, xp);
  k_attn  <<<NSEQ, 256, 0, stream>>>(xp, Wq, rel, stacked, psum, psq);
  k_stats <<<512, 256, 0, stream>>>(psum, psq, gamma, beta, scale, shift);
  k_out   <<<1024, 256, 0, stream>>>(stacked, scale, shift, out);
}